// SpreadTokenFoam_35046933136037
// MI455X (gfx1250) — compile-verified
//
#include <hip/hip_runtime.h>
#include <hip/hip_bf16.h>
#include <math.h>

#define VOCAB 50257
#define DIM 64
#define NB 16
#define SEQ 1024
#define RELAX_C5 0.2373046875f   // 0.75^5 (relax preserves the bubble mean)

typedef float v2f __attribute__((ext_vector_type(2)));
typedef float v8f __attribute__((ext_vector_type(8)));
typedef unsigned int u32x4 __attribute__((ext_vector_type(4)));
typedef int i32x4 __attribute__((ext_vector_type(4)));
typedef int i32x8 __attribute__((ext_vector_type(8)));

// ---------------------------------------------------------------------------
// Kernel A: sequential foam scan. 1 block, 1024 threads (32 waves, one WGP).
// bases (256KB) lives in LDS, transposed + add-swizzled so that the matvec
// reads and the transpose stores are both bank-conflict free.
// ---------------------------------------------------------------------------
__global__ void foam_scan_kernel(const int* __restrict__ tokens,
                                 const float* __restrict__ E,
                                 const float* __restrict__ bases,
                                 const float* __restrict__ mdb_p,
                                 const float* __restrict__ sens_p,
                                 float* __restrict__ out_rho,    // [SEQ,64,64]
                                 float* __restrict__ out_dist,   // [SEQ,16]
                                 float* __restrict__ out_Srho,   // [SEQ]
                                 float* __restrict__ out_nov,    // [SEQ]
                                 float* __restrict__ out_dec,    // [SEQ]
                                 float* __restrict__ ws_en)      // [SEQ,16,64]
{
  extern __shared__ float smem[];
  float* basesT  = smem;                  // 65536 (swizzled transpose)
  float* memS    = basesT + NB*DIM*DIM;   // 1024
  float* eqS     = memS + NB*DIM;         // 1024 (m, then eq in-place)
  float* enS     = eqS + NB*DIM;          // 1024
  float* wenS    = enS + NB*DIM;          // 1024 (p_n * en)
  float* xv      = wenS + NB*DIM;         // 64
  float* mmv     = xv + DIM;              // 64
  float* xwv     = mmv + DIM;             // 64
  float* meanv   = xwv + DIM;             // 64
  float* wred    = meanv + DIM;           // 32
  float* wred2   = wred + 32;             // 32
  float* energyS = wred2 + 32;            // 16
  float* enormS  = energyS + NB;          // 16
  float* pS      = enormS + NB;           // 16
  float* scal    = pS + NB;               // 4

  const int tid  = threadIdx.x;
  const int lane = tid & 31;
  const int wid  = tid >> 5;
  const float mdb  = mdb_p[0];
  const float sens = fabsf(sens_p[0]);

  // Preload bases, transposed: T(n,i,j) at basesT[n*4096 + j*64 + ((i+j)&63)]
  for (int idx = tid; idx < NB*DIM*DIM; idx += blockDim.x) {
    int n = idx >> 12;
    int i = (idx >> 6) & 63;
    int j = idx & 63;
    basesT[n*4096 + j*64 + ((i + j) & 63)] = bases[idx];
  }
  for (int e = tid; e < NB*DIM; e += blockDim.x) memS[e] = 0.f;
  __syncthreads();

  const int n_own = tid >> 6;   // bubble index for [NB,DIM] element tid

  for (int t = 0; t < SEQ; ++t) {
    const int tok = tokens[t];
    if (tid < DIM) {
      xv[tid] = E[(size_t)tok * DIM + tid];
      float s = 0.f;
      #pragma unroll
      for (int n = 0; n < NB; ++n) s += memS[n*DIM + tid];
      mmv[tid] = s * (1.f/NB);
    }
    __syncthreads();
    if (wid == 0) {
      float x0 = xv[lane],  x1 = xv[lane+32];
      float m0 = mmv[lane], m1 = mmv[lane+32];
      float dxm = x0*m0 + x1*m1;
      float dmm = m0*m0 + m1*m1;
      float dxx = x0*x0 + x1*x1;
      for (int m = 16; m >= 1; m >>= 1) {
        dxm += __shfl_xor(dxm, m, 32);
        dmm += __shfl_xor(dmm, m, 32);
        dxx += __shfl_xor(dxx, m, 32);
      }
      if (lane == 0) {
        float mem_norm = sqrtf(dmm) + 1e-10f;
        float x_norm   = sqrtf(dxx) + 1e-10f;
        float nov = (mem_norm > 1e-8f) ? (1.f - dxm / (x_norm * mem_norm)) : 1.f;
        float dec = 1.f / (1.f + expf(-(mdb - sens * nov)));
        scal[0] = dec; scal[1] = nov;
        out_nov[t] = nov; out_dec[t] = dec;
      }
    }
    __syncthreads();
    const float dec = scal[0];
    if (tid < DIM) xwv[tid] = xv[tid] + dec * mmv[tid];
    __syncthreads();
    // m = bases @ x_with_memory  (into eqS)
    {
      const int i = tid & 63;
      const float* bn = basesT + n_own*4096;
      float acc = 0.f;
      #pragma unroll 8
      for (int j = 0; j < DIM; ++j)
        acc += bn[j*64 + ((i + j) & 63)] * xwv[j];
      eqS[tid] = acc;
    }
    __syncthreads();
    if (tid < DIM) {
      float s = 0.f;
      #pragma unroll
      for (int n = 0; n < NB; ++n) s += eqS[n*DIM + tid];
      meanv[tid] = s * (1.f/NB);
    }
    __syncthreads();
    // eq (closed-form relax), per-bubble energy & norm
    {
      float mv   = eqS[tid];
      float mean = meanv[tid & 63];
      float eqv  = mean + RELAX_C5 * (mv - mean);
      float d    = eqv - mv;
      float dsq  = d * d;
      float esq  = eqv * eqv;
      eqS[tid] = eqv;
      for (int m = 16; m >= 1; m >>= 1) {
        dsq += __shfl_xor(dsq, m, 32);
        esq += __shfl_xor(esq, m, 32);
      }
      if (lane == 0) { wred[wid] = dsq; wred2[wid] = esq; }
    }
    __syncthreads();
    if (tid < NB) {
      energyS[tid] = wred[2*tid] + wred[2*tid + 1];
      enormS[tid]  = sqrtf(wred2[2*tid] + wred2[2*tid + 1]);
    }
    __syncthreads();
    if (tid == 0) {
      float mx = -1e30f;
      for (int n = 0; n < NB; ++n) mx = fmaxf(mx, -energyS[n]);
      float ev[NB];
      float z = 0.f;
      for (int n = 0; n < NB; ++n) { ev[n] = expf(-energyS[n] - mx); z += ev[n]; }
      float invz = 1.f / z;
      float s_rho = 0.f;
      for (int n = 0; n < NB; ++n) {
        float pn = ev[n] * invz;
        pS[n] = pn;
        out_dist[t*NB + n] = pn;
        s_rho -= pn * fmaxf(logf(pn), -100.f);
      }
      out_Srho[t] = s_rho;
    }
    __syncthreads();
    {
      float inv = 1.f / (enormS[n_own] + 1e-10f);
      float en  = eqS[tid] * inv;
      enS[tid]  = en;
      wenS[tid] = pS[n_own] * en;
      ws_en[(size_t)t*NB*DIM + tid] = en;
    }
    __syncthreads();
    // rho[i,j] = sum_n wen[n,i] * en[n,j]
    {
      float* rho_t = out_rho + (size_t)t * DIM * DIM;
      #pragma unroll
      for (int k = 0; k < 4; ++k) {
        int e2 = tid + k*1024;
        int i = e2 >> 6, j = e2 & 63;
        float acc = 0.f;
        #pragma unroll
        for (int n = 0; n < NB; ++n)
          acc += wenS[n*DIM + i] * enS[n*DIM + j];
        rho_t[e2] = acc;
      }
    }
    memS[tid] = dec * memS[tid] + (1.f - dec) * eqS[tid];
    __syncthreads();
  }
}

// ---------------------------------------------------------------------------
// Kernel B: logits[t,v] = sum_n p[t,n] * (E[v,:] . en[t,n,:])^2  via WMMA.
// Per wave: one 16x16 fp32 tile, K=64 as 16 chained V_WMMA_F32_16X16X4_F32.
// A sched_barrier separates fragment loads from the WMMA chain so the matrix
// pipe runs without per-step LDS waits. The per-block en slab (8 timesteps,
// 32KB contiguous) is staged into LDS by the Tensor Data Mover.
// ---------------------------------------------------------------------------
#define B_WAVES  8
#define B_ROWS   (B_WAVES * 16)   // 128 vocab rows per block
#define B_TSTEPS 8                // timesteps per block

__global__ void born_logits_kernel(const float* __restrict__ E,
                                   const float* __restrict__ ws_en,
                                   const float* __restrict__ out_dist,
                                   float* __restrict__ out_logits)
{
  __shared__ float enLin[B_TSTEPS * NB * DIM];      // 32KB linear en slab
  __shared__ float enT[DIM * NB];                   // 4KB transposed: enT[j][n]
  __shared__ float red[B_WAVES * 16 * 16];          // 8KB epilogue scratch
  __shared__ float pL[NB];

  const int tid  = threadIdx.x;
  const int lane = tid & 31;
  const int wave = tid >> 5;
  const int half = lane >> 4;
  const int l15  = lane & 15;
  const int vbase = (blockIdx.x * B_WAVES + wave) * 16;
  const int t0 = blockIdx.y * B_TSTEPS;

  // ---- stage the contiguous en slab for t0..t0+7 into LDS via TDM ----
#if __has_builtin(__builtin_amdgcn_tensor_load_to_lds)
  if (wave == 0) {
    const unsigned long long ga =
        (unsigned long long)(const void*)(ws_en + (size_t)t0 * NB * DIM);
    const unsigned lds_off = (unsigned)(size_t)(void*)&enLin[0];
    // D# group0: count=1 | lds_addr | global_addr(57b) | type=2
    u32x4 g0;
    g0.x = 1u;
    g0.y = lds_off;
    g0.z = (unsigned)ga;
    g0.w = ((unsigned)(ga >> 32) & 0x01FFFFFFu) | 0x80000000u;
    // D# group1: data_size=4B, 1-D tile of 8192 elements
    i32x8 g1;
    g1.s0 = 0x00020000;            // workgroup_mask=0, data_size=2 (4 bytes)
    g1.s1 = (int)(8192u << 16);    // tensor_dim0 lo16 in [31:16]
    g1.s2 = 0x00010000;            // tensor_dim0 hi=0, tensor_dim1=1
    g1.s3 = (int)(8192u << 16);    // tile_dim0 = 8192 in [31:16]
    g1.s4 = 1;                     // tile_dim1 = 1, tile_dim2 = 0
    g1.s5 = 8192;                  // tensor_dim0_stride lo32
    g1.s6 = 0;
    g1.s7 = 0;
    i32x4 gz4 = {0, 0, 0, 0};
    i32x8 gz8 = {0, 0, 0, 0, 0, 0, 0, 0};
    __builtin_amdgcn_tensor_load_to_lds(g0, g1, gz4, gz4, gz8, 0);
    __builtin_amdgcn_s_wait_tensorcnt(0);
  }
#else
  for (int e = tid; e < B_TSTEPS * NB * DIM; e += blockDim.x)
    enLin[e] = ws_en[(size_t)t0 * NB * DIM + e];
#endif
  __syncthreads();

  // ---- A fragments: register-resident, reused for all 8 timesteps ----
  int row = vbase + l15;
  if (row >= VOCAB) row = VOCAB - 1;
  const float* Erow = E + (size_t)row * DIM;
  __builtin_prefetch(Erow, 0, 1);          // global_prefetch_b8 (L2-resident E)
  v2f a[16];
  #pragma unroll
  for (int k = 0; k < 16; ++k)             // lanes 0-15: K={4k,4k+1}; 16-31: {4k+2,4k+3}
    a[k] = *(const v2f*)(Erow + k*4 + half*2);

  for (int tt = 0; tt < B_TSTEPS; ++tt) {
    const int t = t0 + tt;
    // transpose this timestep's en into enT[j][n] (conflict-free frag reads)
    for (int e = tid; e < NB * DIM; e += blockDim.x) {
      int n = e >> 6, j = e & 63;
      enT[j * NB + n] = enLin[tt * NB * DIM + e];
    }
    if (tid < NB) pL[tid] = out_dist[t * NB + tid];
    __syncthreads();

    // B fragments fully into registers, then an uninterrupted WMMA chain.
    v2f b[16];
    #pragma unroll
    for (int k = 0; k < 16; ++k) {
      int col = k*4 + half*2;              // B[K=col][N=l15] = en[n=l15][j=col]
      b[k].x = enT[col * NB + l15];
      b[k].y = enT[(col + 1) * NB + l15];
    }
#if __has_builtin(__builtin_amdgcn_sched_barrier)
    __builtin_amdgcn_sched_barrier(0);     // all 16 frag loads issue first,
#endif                                     // then one wait, then 16 WMMAs
    v8f c = {};
    #pragma unroll
    for (int k = 0; k < 16; ++k)
      c = __builtin_amdgcn_wmma_f32_16x16x4_f32(
              false, a[k], false, b[k], (short)0, c, false, false);

    // Epilogue: logits[v=M] = sum_N p[N]*C[M,N]^2 via per-wave LDS tile.
    // C layout: VGPR r -> M=r (lanes 0-15), M=r+8 (lanes 16-31), N=lane%16.
    const float pl = pL[l15];
    float* rw = red + wave * 256;
    #pragma unroll
    for (int r = 0; r < 8; ++r) {
      float cv = c[r];
      rw[(r + 8*half) * 16 + l15] = pl * cv * cv;
    }
    if (l15 < 8) {
      int m = half * 8 + l15;
      const float4* rp = (const float4*)(rw + m * 16);
      float4 q0 = rp[0], q1 = rp[1], q2 = rp[2], q3 = rp[3];
      float val = (((q0.x + q0.y) + (q0.z + q0.w)) + ((q1.x + q1.y) + (q1.z + q1.w)))
                + (((q2.x + q2.y) + (q2.z + q2.w)) + ((q3.x + q3.y) + (q3.z + q3.w)));
      int v = vbase + m;
      if (v < VOCAB) out_logits[(size_t)t * VOCAB + v] = val;
    }
    __syncthreads();   // protect enT/red before next timestep
  }
}

// ---------------------------------------------------------------------------
// Kernel C: per-timestep vocab softmax + entropy. One 256-thread block per t.
// ---------------------------------------------------------------------------
__global__ void softmax_entropy_kernel(const float* __restrict__ logits_all,
                                       float* __restrict__ probs_all,
                                       const float* __restrict__ S_rho,
                                       float* __restrict__ H_out,
                                       float* __restrict__ F_out)
{
  const int t = blockIdx.x;
  const float* l = logits_all + (size_t)t * VOCAB;
  float* pr = probs_all + (size_t)t * VOCAB;
  __shared__ float red[8];
  const int tid = threadIdx.x;
  const int lane = tid & 31, wid = tid >> 5;

  float mx = -1e30f;
  for (int v = tid; v < VOCAB; v += blockDim.x) mx = fmaxf(mx, l[v]);
  for (int m = 16; m >= 1; m >>= 1) mx = fmaxf(mx, __shfl_xor(mx, m, 32));
  if (lane == 0) red[wid] = mx;
  __syncthreads();
  if (tid == 0) { float r = red[0]; for (int w = 1; w < 8; ++w) r = fmaxf(r, red[w]); red[0] = r; }
  __syncthreads();
  mx = red[0];
  __syncthreads();

  float z = 0.f;
  for (int v = tid; v < VOCAB; v += blockDim.x) z += expf(l[v] - mx);
  for (int m = 16; m >= 1; m >>= 1) z += __shfl_xor(z, m, 32);
  if (lane == 0) red[wid] = z;
  __syncthreads();
  if (tid == 0) { float r = 0.f; for (int w = 0; w < 8; ++w) r += red[w]; red[0] = r; }
  __syncthreads();
  z = red[0];
  __syncthreads();
  const float logZ = logf(z);

  float h = 0.f;
  for (int v = tid; v < VOCAB; v += blockDim.x) {
    float lp = l[v] - mx - logZ;
    float pv = expf(lp);
    pr[v] = pv;
    h -= pv * fmaxf(lp, -100.f);
  }
  for (int m = 16; m >= 1; m >>= 1) h += __shfl_xor(h, m, 32);
  if (lane == 0) red[wid] = h;
  __syncthreads();
  if (tid == 0) {
    float r = 0.f; for (int w = 0; w < 8; ++w) r += red[w];
    H_out[t] = r;
    F_out[t] = r - S_rho[t];
  }
}

// ---------------------------------------------------------------------------
extern "C" void kernel_launch(void* const* d_in, const int* in_sizes, int n_in,
                              void* d_out, int out_size, void* d_ws, size_t ws_size,
                              hipStream_t stream) {
  const int*   tokens = (const int*)  d_in[0];
  const float* E      = (const float*)d_in[1];
  const float* bases  = (const float*)d_in[2];
  const float* mdb    = (const float*)d_in[3];
  const float* sens   = (const float*)d_in[4];

  float* out = (float*)d_out;
  const size_t L = (size_t)SEQ * VOCAB;
  float* o_logits = out;                               // [SEQ,VOCAB]
  float* o_probs  = out + L;                           // [SEQ,VOCAB]
  float* o_rho    = out + 2*L;                         // [SEQ,64,64]
  float* o_dist   = o_rho  + (size_t)SEQ*DIM*DIM;      // [SEQ,16]
  float* o_Srho   = o_dist + (size_t)SEQ*NB;           // [SEQ]
  float* o_H      = o_Srho + SEQ;                      // [SEQ]
  float* o_F      = o_H    + SEQ;                      // [SEQ]
  float* o_nov    = o_F    + SEQ;                      // [SEQ]
  float* o_dec    = o_nov  + SEQ;                      // [SEQ]

  float* ws_en = (float*)d_ws;                         // SEQ*NB*DIM floats (4MB)

  const size_t smemA = (size_t)(NB*DIM*DIM + 4*NB*DIM + 4*DIM + 64 + 3*NB + 4)
                       * sizeof(float);                // ~280KB (<= 320KB/WGP)
  hipLaunchKernelGGL(foam_scan_kernel, dim3(1), dim3(1024), smemA, stream,
                     tokens, E, bases, mdb, sens,
                     o_rho, o_dist, o_Srho, o_nov, o_dec, ws_en);

  dim3 gB((VOCAB + B_ROWS - 1) / B_ROWS, SEQ / B_TSTEPS);
  hipLaunchKernelGGL(born_logits_kernel, gB, dim3(B_WAVES * 32), 0, stream,
                     E, ws_en, o_dist, o_logits);

  hipLaunchKernelGGL(softmax_entropy_kernel, dim3(SEQ), dim3(256), 0, stream,
                     o_logits, o_probs, o_Srho, o_H, o_F);
}